// MultiheadAttention_48361331753823
// MI455X (gfx1250) — compile-verified
//
#include <hip/hip_runtime.h>
#include <hip/hip_bf16.h>

// ---------------------------------------------------------------------------
// MHA forward for MI455X (gfx1250), f32 end-to-end via V_WMMA_F32_16X16X4_F32.
//   B=4, L=2048, D=512, H=8, Dh=64
// ---------------------------------------------------------------------------

typedef __attribute__((ext_vector_type(2))) float v2f;
typedef __attribute__((ext_vector_type(8))) float v8f;

#define Bsz 4
#define Lsz 2048
#define Dm  512
#define Hn  8
#define Dh  64
#define SROW 2049                 // 2048 + 1 pad -> conflict-free column access
#define SCALE 0.04419417382415922f // 1/sqrt(512)

__device__ __forceinline__ v8f wmma_f32(v2f a, v2f b, v8f c) {
  // (neg_a, A, neg_b, B, c_mod, C, reuse_a, reuse_b)
  return __builtin_amdgcn_wmma_f32_16x16x4_f32(false, a, false, b, (short)0, c,
                                               false, false);
}

__device__ __forceinline__ v8f vzero8() {
  v8f z;
#pragma unroll
  for (int i = 0; i < 8; ++i) z[i] = 0.0f;
  return z;
}

// ---------------------------------------------------------------------------
// Y[M,512] = X[M,512] @ W[512,512]^T + bias[512]
// One wave computes a 16x64 output tile (4 WMMA accumulators), K-loop step 4.
// A operand: lane m=l16 holds X[m][k0+2h .. +1]  -> one b64 load.
// B operand: lane n=l16 holds W[n][k0+2h .. +1]  -> one b64 load (row-major K).
// ---------------------------------------------------------------------------
__global__ void mha_gemm_bias(const float* __restrict__ X,
                              const float* __restrict__ W,
                              const float* __restrict__ bias,
                              float* __restrict__ Y) {
  const int lane = threadIdx.x & 31;
  const int l16  = lane & 15;
  const int half = lane >> 4;

  const int wid = blockIdx.x * (blockDim.x >> 5) + (threadIdx.x >> 5);
  const int tm  = wid >> 3;        // 512 row tiles of 16 (M = 8192)
  const int tn  = wid & 7;         // 8 col tiles of 64 (N = 512)
  const int m0  = tm * 16;
  const int n0  = tn * 64;

  const float* arow = X + (m0 + l16) * Dm + 2 * half;
  const float* brow0 = W + (n0 +  0 + l16) * Dm + 2 * half;
  const float* brow1 = W + (n0 + 16 + l16) * Dm + 2 * half;
  const float* brow2 = W + (n0 + 32 + l16) * Dm + 2 * half;
  const float* brow3 = W + (n0 + 48 + l16) * Dm + 2 * half;

  v8f acc0 = vzero8(), acc1 = vzero8(), acc2 = vzero8(), acc3 = vzero8();

#pragma unroll 4
  for (int k0 = 0; k0 < Dm; k0 += 4) {
    v2f a  = *(const v2f*)(arow  + k0);
    v2f b0 = *(const v2f*)(brow0 + k0);
    v2f b1 = *(const v2f*)(brow1 + k0);
    v2f b2 = *(const v2f*)(brow2 + k0);
    v2f b3 = *(const v2f*)(brow3 + k0);
    acc0 = wmma_f32(a, b0, acc0);
    acc1 = wmma_f32(a, b1, acc1);
    acc2 = wmma_f32(a, b2, acc2);
    acc3 = wmma_f32(a, b3, acc3);
  }

  v8f accs[4] = {acc0, acc1, acc2, acc3};
#pragma unroll
  for (int s = 0; s < 4; ++s) {
    const int n  = n0 + s * 16 + l16;
    const float bb = bias[n];
#pragma unroll
    for (int r = 0; r < 8; ++r) {
      const int m = m0 + r + 8 * half;
      Y[m * Dm + n] = accs[s][r] + bb;
    }
  }
}

// ---------------------------------------------------------------------------
// Fused attention: one block = (b, h, 16-query block).
//  Phase 1: S[16,2048] = qp @ kp^T * scale, masked -> LDS (waves split keys)
//  Phase 2: row softmax stats (max, sum of exp; exp stored back to LDS)
//  Phase 3: coalesced write of normalized attn rows to global (the 512 MiB)
//  Phase 4: ctx[16,64] = attn @ vp via WMMA, K split across waves, ds_add_f32
// ---------------------------------------------------------------------------
extern "C" __global__ void __launch_bounds__(256)
mha_attention(const float* __restrict__ qp, const float* __restrict__ kp,
              const float* __restrict__ vp, const int* __restrict__ mask,
              float* __restrict__ attn, float* __restrict__ ctx) {
  extern __shared__ float smem[];
  float* S      = smem;                    // 16 * 2049
  float* red    = S + 16 * SROW;           // 256
  float* rowmax = red + 256;               // 16
  float* rowinv = rowmax + 16;             // 16
  float* ctxs   = rowinv + 16;             // 16 * 64

  const int tid  = threadIdx.x;
  const int lane = tid & 31;
  const int l16  = lane & 15;
  const int half = lane >> 4;
  const int wv   = tid >> 5;               // wave 0..7

  const int qblk = blockIdx.x & 127;
  const int h    = (blockIdx.x >> 7) & 7;
  const int b    = blockIdx.x >> 10;
  const int q0   = qblk * 16;

  // zero ctx accumulator
  for (int i = tid; i < 16 * Dh; i += 256) ctxs[i] = 0.0f;

  // ---- Phase 1: scores for this wave's 256 keys ----
  const float* qrow = qp + (b * Lsz + q0 + l16) * Dm + h * Dh + 2 * half;
  v2f areg[16];
#pragma unroll
  for (int ks = 0; ks < 16; ++ks) areg[ks] = *(const v2f*)(qrow + ks * 4);

  const int* mbase = mask + (b * Lsz + q0) * Lsz;
  for (int t = 0; t < 16; ++t) {
    const int nt  = wv * 16 + t;           // key tile index
    const int key = nt * 16 + l16;
    const float* krow = kp + (b * Lsz + key) * Dm + h * Dh + 2 * half;
    v8f c = vzero8();
#pragma unroll
    for (int ks = 0; ks < 16; ++ks) {
      v2f bv = *(const v2f*)(krow + ks * 4);
      c = wmma_f32(areg[ks], bv, c);
    }
    const int col = nt * 16 + l16;
    const int* mcol = mbase + col;
#pragma unroll
    for (int r = 0; r < 8; ++r) {
      const int m = r + 8 * half;
      float sv = c[r] * SCALE;
      if (mcol[m * Lsz] == 0) sv = -1e9f;
      S[m * SROW + col] = sv;
    }
  }
  __syncthreads();

  // ---- Phase 2: softmax stats (16 threads per row, stride-16 columns) ----
  const int r = tid >> 4;
  const int j = tid & 15;
  float lmax = -3.0e38f;
  for (int c = j; c < Lsz; c += 16) lmax = fmaxf(lmax, S[r * SROW + c]);
  red[r * 16 + j] = lmax;
  __syncthreads();
  if (j == 0) {
    float mx = red[r * 16];
#pragma unroll
    for (int t = 1; t < 16; ++t) mx = fmaxf(mx, red[r * 16 + t]);
    rowmax[r] = mx;
  }
  __syncthreads();
  const float mx = rowmax[r];
  float lsum = 0.0f;
  for (int c = j; c < Lsz; c += 16) {
    float e = __expf(S[r * SROW + c] - mx);
    S[r * SROW + c] = e;
    lsum += e;
  }
  red[r * 16 + j] = lsum;
  __syncthreads();
  if (j == 0) {
    float sm = 0.0f;
#pragma unroll
    for (int t = 0; t < 16; ++t) sm += red[r * 16 + t];
    rowinv[r] = 1.0f / sm;
  }
  __syncthreads();

  // ---- Phase 3: write normalized attention (contiguous, coalesced) ----
  float* attnb = attn + ((size_t)((b * Hn + h) * Lsz + q0)) * Lsz;
  for (int idx = tid; idx < 16 * Lsz; idx += 256) {
    const int rr = idx >> 11;
    const int cc = idx & 2047;
    attnb[idx] = S[rr * SROW + cc] * rowinv[rr];
  }

  // ---- Phase 4: ctx = attn @ vp  (wave -> one 16-wide d tile, half of K) ----
  const int d0    = (wv & 3) * 16;
  const int kbase = (wv >> 2) * 1024;
  const int d     = d0 + l16;
  const float* vcol = vp + (size_t)(b * Lsz) * Dm + h * Dh + d;
  v8f acc = vzero8();
#pragma unroll 4
  for (int k0 = kbase; k0 < kbase + 1024; k0 += 4) {
    const int kk = k0 + 2 * half;
    v2f a, bv;
    a.x  = S[l16 * SROW + kk];
    a.y  = S[l16 * SROW + kk + 1];
    bv.x = vcol[(size_t)kk * Dm];
    bv.y = vcol[(size_t)(kk + 1) * Dm];
    acc = wmma_f32(a, bv, acc);
  }
#pragma unroll
  for (int rr = 0; rr < 8; ++rr) {
    const int m = rr + 8 * half;
    atomicAdd(&ctxs[m * Dh + d], acc[rr] * rowinv[m]);
  }
  __syncthreads();

  // ---- Phase 5: store ctx tile in [B, L, H*Dh] layout ----
  for (int idx = tid; idx < 16 * Dh; idx += 256) {
    const int m = idx >> 6;
    const int dd = idx & 63;
    ctx[(b * Lsz + q0 + m) * Dm + h * Dh + dd] = ctxs[idx];
  }
}

// ---------------------------------------------------------------------------
extern "C" void kernel_launch(void* const* d_in, const int* in_sizes, int n_in,
                              void* d_out, int out_size, void* d_ws, size_t ws_size,
                              hipStream_t stream) {
  const float* q    = (const float*)d_in[0];
  const float* k    = (const float*)d_in[1];
  const float* v    = (const float*)d_in[2];
  const int*   mask = (const int*)d_in[3];
  const float* wq   = (const float*)d_in[4];
  const float* bq   = (const float*)d_in[5];
  const float* wo   = (const float*)d_in[6];
  const float* bo   = (const float*)d_in[7];

  float* out  = (float*)d_out;
  float* attn = out + (size_t)Bsz * Lsz * Dm;          // outputs: (out, attn)

  const size_t NP = (size_t)Bsz * Lsz * Dm;            // 4.19M floats each
  float* ws  = (float*)d_ws;
  float* qp  = ws;
  float* kp  = ws + NP;
  float* vp  = ws + 2 * NP;
  float* ctx = ws + 3 * NP;

  // Projections (reference bug: Wq/bq applied to q, k, AND v)
  mha_gemm_bias<<<512, 256, 0, stream>>>(q, wq, bq, qp);
  mha_gemm_bias<<<512, 256, 0, stream>>>(k, wq, bq, kp);
  mha_gemm_bias<<<512, 256, 0, stream>>>(v, wq, bq, vp);

  // Fused scores + softmax + attn-write + ctx
  const int smem_bytes = (16 * SROW + 256 + 16 + 16 + 16 * Dh) * (int)sizeof(float);
  (void)hipFuncSetAttribute((const void*)mha_attention,
                            hipFuncAttributeMaxDynamicSharedMemorySize, smem_bytes);
  mha_attention<<<Bsz * Hn * (Lsz / 16), 256, smem_bytes, stream>>>(
      qp, kp, vp, mask, attn, ctx);

  // Output projection
  mha_gemm_bias<<<512, 256, 0, stream>>>(ctx, wo, bo, out);
}